// GNN_23897198035179
// MI455X (gfx1250) — compile-verified
//
#include <hip/hip_runtime.h>
#include <math.h>

// fp32 WMMA: D(16x16) = A(16x4) * B(4x16) + C  — exact fp32 on the matrix pipe
typedef __attribute__((ext_vector_type(2))) float v2f;
typedef __attribute__((ext_vector_type(8))) float v8f;
typedef __attribute__((ext_vector_type(4))) int   v4i;

#define WMMA_F32(a, b, c) \
    __builtin_amdgcn_wmma_f32_16x16x4_f32(false, (a), false, (b), (short)0, (c), false, false)

// ---- CDNA5 async global->LDS (ASYNCcnt) with portable fallback --------------
#if defined(__has_builtin)
#if __has_builtin(__builtin_amdgcn_global_load_async_to_lds_b128)
#define HAVE_ASYNC_LDS 1
#endif
#endif

#if defined(HAVE_ASYNC_LDS)
typedef __attribute__((address_space(1))) v4i* gv4i_ptr;   // global
typedef __attribute__((address_space(3))) v4i* lv4i_ptr;   // LDS
#endif

template <int NCNT>
static __device__ __forceinline__ void wait_asynccnt() {
#if defined(HAVE_ASYNC_LDS)
#if defined(__has_builtin)
#if __has_builtin(__builtin_amdgcn_s_wait_asynccnt)
    __builtin_amdgcn_s_wait_asynccnt(NCNT);
#else
    asm volatile("s_wait_asynccnt %0" ::"i"(NCNT));
#endif
#endif
#endif
}

static __device__ __forceinline__ float warp_sum(float v) {
    // wave32 butterfly reduction; all lanes end with the total
    for (int m = 16; m > 0; m >>= 1) v += __shfl_xor(v, m, 32);
    return v;
}

static __device__ __forceinline__ void atomicAddF(float* p, float v) {
    unsafeAtomicAdd(p, v);   // lowers to global_atomic_add_f32 (no CAS loop)
}

__global__ void zero_kernel(float* __restrict__ p, int n) {
    int i = blockIdx.x * blockDim.x + threadIdx.x;
    if (i < n) p[i] = 0.0f;
}

// ---------------------------------------------------------------------------
// out[N, NC] = act(A[N, K]) @ W[K, NC] + bias     (act = relu if RELU)
// Block = 128 threads (4 waves). Each wave owns a 16x16 output tile;
// block covers 16 rows x 64 cols. A tile staged in padded LDS.
// ---------------------------------------------------------------------------
template <int K, int NC, bool RELU>
__global__ __launch_bounds__(128) void gemm_bias_kernel(
    const float* __restrict__ A, const float* __restrict__ W,
    const float* __restrict__ bias, float* __restrict__ out) {
    constexpr int LK = K + 4;                 // pad: stride ≡ 4 (mod 64 banks)
    __shared__ float As[16 * LK];
    const int lane = threadIdx.x & 31;
    const int wave = threadIdx.x >> 5;
    const int hh   = lane >> 4;               // lane half (0/1)
    const int l16  = lane & 15;
    const int koff = hh * 2;
    const int row0 = blockIdx.y * 16;
    const int colb = blockIdx.x * 64 + wave * 16;
    const int col  = colb + l16;

    for (int idx = threadIdx.x; idx < 16 * K; idx += 128) {
        int r = idx / K, k = idx % K;         // K is a power of two -> shifts
        float v = A[(size_t)(row0 + r) * K + k];
        if (RELU) v = fmaxf(v, 0.0f);
        As[r * LK + k] = v;
    }
    __syncthreads();

    v8f c;
    #pragma unroll
    for (int q = 0; q < 8; ++q) c[q] = 0.0f;

    #pragma unroll 8
    for (int kb = 0; kb < K; kb += 4) {
        v2f a, b;
        a.x = As[l16 * LK + kb + koff];       // A frag: row=l16, K=kb+koff(+1)
        a.y = As[l16 * LK + kb + koff + 1];
        b.x = W[(size_t)(kb + koff) * NC + col];      // B frag: W[k][col]
        b.y = W[(size_t)(kb + koff + 1) * NC + col];
        c = WMMA_F32(a, b, c);
    }

    const float bv = bias[col];
    #pragma unroll
    for (int v = 0; v < 8; ++v) {
        int r = v + hh * 8;                   // D layout: row = v + 8*half
        out[(size_t)(row0 + r) * NC + col] = c[v] + bv;
    }
}

// ---------------------------------------------------------------------------
// Per-node stats: al = h·aw[:dh], ar = h·aw[dh:], nrm = ||h||.  Wave per node.
// ---------------------------------------------------------------------------
__global__ __launch_bounds__(128) void stats_kernel(
    const float* __restrict__ h, const float* __restrict__ aw,
    float* __restrict__ al, float* __restrict__ ar, float* __restrict__ nrm, int dh) {
    const int lane = threadIdx.x & 31;
    const int node = blockIdx.x * 4 + (threadIdx.x >> 5);
    const float* hp = h + (size_t)node * dh;
    float sl = 0.f, sr = 0.f, sq = 0.f;
    for (int c = lane; c < dh; c += 32) {
        float v = hp[c];
        sl += v * aw[c];
        sr += v * aw[dh + c];
        sq += v * v;
    }
    sl = warp_sum(sl); sr = warp_sum(sr); sq = warp_sum(sq);
    if (lane == 0) { al[node] = sl; ar[node] = sr; nrm[node] = sqrtf(sq); }
}

// ---------------------------------------------------------------------------
// Edge pass 1: e = exp(leaky(al[s]+ar[d]+ab)); denom[s] += e   (thread/edge)
// ---------------------------------------------------------------------------
__global__ void edge1_kernel(
    const long long* __restrict__ src, const long long* __restrict__ dst,
    const float* __restrict__ al, const float* __restrict__ ar,
    const float* __restrict__ abp, float* __restrict__ evals,
    float* __restrict__ denom, int E) {
    int e = blockIdx.x * blockDim.x + threadIdx.x;
    if (e >= E) return;
    int s = (int)src[e], d = (int)dst[e];
    float x  = al[s] + ar[d] + abp[0];
    float lx = x > 0.f ? x : 0.01f * x;
    float ev = __expf(lx);
    evals[e] = ev;
    atomicAddF(&denom[s], ev);
}

// ---------------------------------------------------------------------------
// Edge pass 2: acc[s] += (e/denom[s]) * h[d]    (wave per edge, lanes over dh)
// ---------------------------------------------------------------------------
__global__ __launch_bounds__(128) void edge2_kernel(
    const long long* __restrict__ src, const long long* __restrict__ dst,
    const float* __restrict__ evals, const float* __restrict__ denom,
    const float* __restrict__ h, float* __restrict__ acc, int dh) {
    const int lane = threadIdx.x & 31;
    const int e = blockIdx.x * 4 + (threadIdx.x >> 5);
    int s = (int)src[e], d = (int)dst[e];
    float a = evals[e] / denom[s];
    const float* hd = h + (size_t)d * dh;
    float* ap = acc + (size_t)s * dh;
    for (int c = lane; c < dh; c += 32) atomicAddF(&ap[c], a * hd[c]);
}

// ---------------------------------------------------------------------------
// Fused dense-attention (flash style), double-buffered J tiles via CDNA5
// async global->LDS (ASYNCcnt). Block = 4 waves, one 16-row tile.
//   per 16-col tile J:
//     C_tile  = h_I @ h_Jᵀ   (waves split dh into 4 K-chunks, WMMA, LDS reduce)
//     B_tile  = exp(thresh(C/(|hi||hj|)) * leaky(al_i+ar_j+ab))
//     acc    += B_tile @ h_J  (waves split dh output columns, WMMA)
//   out = h + 0.5*(acc + edge_acc)    [== 0.5*(h_a + h1) of the reference]
// B (4096x4096, 64MB) is never materialized; h tiles live in L2/LDS.
// ---------------------------------------------------------------------------
template <int DH>
__global__ __launch_bounds__(128) void dense_attn_kernel(
    const float* __restrict__ h, const float* __restrict__ al,
    const float* __restrict__ ar, const float* __restrict__ abp,
    const float* __restrict__ nrm, const float* __restrict__ edge_acc,
    float* __restrict__ out, int N) {
    constexpr int LDH   = DH + 4;             // LDS stride ≡ 4 (mod 64 banks)
    constexpr int CHUNK = DH / 4;             // K-chunk / col-chunk per wave
    constexpr int NSUB  = CHUNK / 16;         // v8f accumulators per wave
    constexpr int F4PT  = (16 * DH) / (128 * 4);  // float4 async ops per thread
    __shared__ float hI[16 * LDH];
    __shared__ float hJ[2][16 * LDH];         // double buffer
    __shared__ float Cp[4][16][16];           // per-wave partial C tiles
    __shared__ float Bt[16][20];              // padded to avoid conflicts
    __shared__ float alI[16], nI[16], arJ[16], nJ[16];

    const int lane = threadIdx.x & 31;
    const int wave = threadIdx.x >> 5;
    const int hh   = lane >> 4;
    const int l16  = lane & 15;
    const int koff = hh * 2;
    const int i0   = blockIdx.x * 16;
    const float ab = abp[0];

    // Issue J-tile fill: async global->LDS when available (no VGPR staging),
    // otherwise plain vectorized staging into the same double buffer.
    auto fill_tile = [&](int buf, int j0) {
        #pragma unroll
        for (int i = 0; i < F4PT; ++i) {
            int f   = (threadIdx.x + i * 128) * 4;   // flat float index
            int row = f / DH;                        // DH pow2 -> shifts
            int col = f % DH;
            const float* gp = h + (size_t)(j0 + row) * DH + col;
            float* lp = &hJ[buf][row * LDH + col];
#if defined(HAVE_ASYNC_LDS)
            __builtin_amdgcn_global_load_async_to_lds_b128(
                (gv4i_ptr)gp, (lv4i_ptr)lp, 0, 0);
#else
            *(float4*)lp = *(const float4*)gp;
#endif
        }
    };

    fill_tile(0, 0);                          // prefetch first J tile
    for (int idx = threadIdx.x; idx < 16 * DH; idx += 128) {
        int r = idx / DH, k = idx % DH;
        hI[r * LDH + k] = h[(size_t)(i0 + r) * DH + k];
    }
    if (threadIdx.x < 16) {
        alI[threadIdx.x] = al[i0 + threadIdx.x];
        nI[threadIdx.x]  = nrm[i0 + threadIdx.x];
    }

    v8f acc[NSUB];
    #pragma unroll
    for (int s = 0; s < NSUB; ++s)
        #pragma unroll
        for (int q = 0; q < 8; ++q) acc[s][q] = 0.0f;

    int cur = 0;
    for (int j0 = 0; j0 < N; j0 += 16) {
        if (j0 + 16 < N) {
            fill_tile(cur ^ 1, j0 + 16);      // keep next tile in flight
            wait_asynccnt<F4PT>();            // current tile complete
        } else {
            wait_asynccnt<0>();
        }
        if (threadIdx.x < 16) {
            arJ[threadIdx.x] = ar[j0 + threadIdx.x];
            nJ[threadIdx.x]  = nrm[j0 + threadIdx.x];
        }
        __syncthreads();
        const float* __restrict__ hj = hJ[cur];

        // ---- C partial over this wave's K chunk (h_I @ h_Jᵀ) ----
        v8f cp;
        #pragma unroll
        for (int q = 0; q < 8; ++q) cp[q] = 0.0f;
        const int ks = wave * CHUNK;
        #pragma unroll
        for (int kb = 0; kb < CHUNK; kb += 4) {
            v2f a, b;
            a.x = hI[l16 * LDH + ks + kb + koff];     // A: row=l16
            a.y = hI[l16 * LDH + ks + kb + koff + 1];
            b.x = hj[l16 * LDH + ks + kb + koff];     // B = h_Jᵀ: same pattern
            b.y = hj[l16 * LDH + ks + kb + koff + 1];
            cp = WMMA_F32(a, b, cp);
        }
        #pragma unroll
        for (int v = 0; v < 8; ++v) Cp[wave][v + hh * 8][l16] = cp[v];
        __syncthreads();

        // ---- elementwise: reduce partials, cosine-threshold, exp ----
        #pragma unroll
        for (int t = 0; t < 2; ++t) {
            int e = threadIdx.x + t * 128;            // 256 elems / 128 thr
            int m = e >> 4, n = e & 15;
            float raw = Cp[0][m][n] + Cp[1][m][n] + Cp[2][m][n] + Cp[3][m][n];
            float cs  = raw / (nI[m] * nJ[n]);
            float cth = cs > 0.36f ? cs : 0.0f;       // THRESH = 0.6^2
            float s   = alI[m] + arJ[n] + ab;
            float ls  = s > 0.f ? s : 0.01f * s;      // leaky_relu(0.01)
            Bt[m][n]  = __expf(cth * ls);
        }
        __syncthreads();

        // ---- acc += B_tile @ h_J (this wave's output columns) ----
        const int cb = wave * CHUNK;
        #pragma unroll
        for (int sub = 0; sub < NSUB; ++sub) {
            const int colb = cb + sub * 16;
            #pragma unroll
            for (int kb = 0; kb < 16; kb += 4) {
                v2f a, b;
                a.x = Bt[l16][kb + koff];                       // A = B_tile
                a.y = Bt[l16][kb + koff + 1];
                b.x = hj[(kb + koff) * LDH + colb + l16];       // B = h_J
                b.y = hj[(kb + koff + 1) * LDH + colb + l16];
                acc[sub] = WMMA_F32(a, b, acc[sub]);
            }
        }
        __syncthreads();                      // protect buffers before reuse
        cur ^= 1;
    }

    // ---- epilogue: out = h + 0.5*(dense + edge) ----
    const int cb = wave * CHUNK;
    #pragma unroll
    for (int sub = 0; sub < NSUB; ++sub) {
        int col = cb + sub * 16 + l16;
        #pragma unroll
        for (int v = 0; v < 8; ++v) {
            int r = v + hh * 8;
            size_t gi = (size_t)(i0 + r) * DH + col;
            out[gi] = hI[r * LDH + col] + 0.5f * (acc[sub][v] + edge_acc[gi]);
        }
    }
}

// Row L2-normalize (wave per node)
__global__ __launch_bounds__(128) void normalize_kernel(
    const float* __restrict__ h, float* __restrict__ out, int dh) {
    const int lane = threadIdx.x & 31;
    const int node = blockIdx.x * 4 + (threadIdx.x >> 5);
    const float* hp = h + (size_t)node * dh;
    float sq = 0.f;
    for (int c = lane; c < dh; c += 32) { float v = hp[c]; sq += v * v; }
    sq = warp_sum(sq);
    float inv = 1.0f / fmaxf(sqrtf(sq), 1e-12f);
    for (int c = lane; c < dh; c += 32)
        out[(size_t)node * dh + c] = hp[c] * inv;
}

extern "C" void kernel_launch(void* const* d_in, const int* in_sizes, int n_in,
                              void* d_out, int out_size, void* d_ws, size_t ws_size,
                              hipStream_t stream) {
    (void)in_sizes; (void)n_in; (void)out_size; (void)ws_size;
    constexpr int N = 4096, E = 131072, D = 128, H1 = 256, H2 = 128, H3 = 64;

    const float*     X    = (const float*)d_in[0];
    const long long* ei   = (const long long*)d_in[1];   // int64 [2, E]
    const float*     W1   = (const float*)d_in[2];
    const float*     b1   = (const float*)d_in[3];
    const float*     a1w  = (const float*)d_in[4];
    const float*     a1b  = (const float*)d_in[5];
    const float*     W2   = (const float*)d_in[6];
    const float*     b2   = (const float*)d_in[7];
    const float*     a2w  = (const float*)d_in[8];
    const float*     a2b  = (const float*)d_in[9];
    const float*     W3   = (const float*)d_in[10];
    const float*     b3   = (const float*)d_in[11];
    const long long* esrc = ei;
    const long long* edst = ei + E;

    float* p     = (float*)d_ws;
    float* h1    = p; p += N * H1;
    float* t1    = p; p += N * H1;
    float* h2    = p; p += N * H2;
    float* t2    = p; p += N * H2;
    float* h3    = p; p += N * H3;
    float* al    = p; p += N;
    float* ar    = p; p += N;
    float* nrm   = p; p += N;
    float* denom = p; p += N;
    float* evals = p; p += E;
    float* eacc  = p; p += N * H1;   // sized for the largest dh

    // ---- layer 1: h1 = X @ W1 + b1 ; attention(dh=256) -> t1 ----
    gemm_bias_kernel<D, H1, false><<<dim3(H1 / 64, N / 16), 128, 0, stream>>>(X, W1, b1, h1);
    stats_kernel<<<N / 4, 128, 0, stream>>>(h1, a1w, al, ar, nrm, H1);
    zero_kernel<<<(N + 255) / 256, 256, 0, stream>>>(denom, N);
    zero_kernel<<<(N * H1 + 255) / 256, 256, 0, stream>>>(eacc, N * H1);
    edge1_kernel<<<E / 256, 256, 0, stream>>>(esrc, edst, al, ar, a1b, evals, denom, E);
    edge2_kernel<<<E / 4, 128, 0, stream>>>(esrc, edst, evals, denom, h1, eacc, H1);
    dense_attn_kernel<H1><<<N / 16, 128, 0, stream>>>(h1, al, ar, a1b, nrm, eacc, t1, N);

    // ---- layer 2: h2 = relu(t1) @ W2 + b2 ; attention(dh=128) -> t2 ----
    gemm_bias_kernel<H1, H2, true><<<dim3(H2 / 64, N / 16), 128, 0, stream>>>(t1, W2, b2, h2);
    stats_kernel<<<N / 4, 128, 0, stream>>>(h2, a2w, al, ar, nrm, H2);
    zero_kernel<<<(N + 255) / 256, 256, 0, stream>>>(denom, N);
    zero_kernel<<<(N * H2 + 255) / 256, 256, 0, stream>>>(eacc, N * H2);
    edge1_kernel<<<E / 256, 256, 0, stream>>>(esrc, edst, al, ar, a2b, evals, denom, E);
    edge2_kernel<<<E / 4, 128, 0, stream>>>(esrc, edst, evals, denom, h2, eacc, H2);
    dense_attn_kernel<H2><<<N / 16, 128, 0, stream>>>(h2, al, ar, a2b, nrm, eacc, t2, N);

    // ---- layer 3: h3 = relu(t2) @ W3 + b3 ; row-normalize -> d_out ----
    gemm_bias_kernel<H2, H3, true><<<dim3(1, N / 16), 128, 0, stream>>>(t2, W3, b3, h3);
    normalize_kernel<<<N / 4, 128, 0, stream>>>(h3, (float*)d_out, H3);
}